// SparseLinear_13176959664756
// MI455X (gfx1250) — compile-verified
//
#include <hip/hip_runtime.h>

// SpMM (COO, sorted rows) * dense fp32 weight + bias, mapped onto
// V_WMMA_F32_16X16X4_F32 on gfx1250 (wave32).
//
// out[4096 x 256] = sparse(4096 x 32000) @ weight(32000 x 256) + bias
//
// 256 workgroups; each owns a 16-row output slab whose nonzeros form a
// contiguous COO range (rows sorted) found via binary search. Nonzeros feed
// the K-dim of fp32 16x16x4 WMMAs: A = one-hot expansion (row match -> val),
// B = gathered weight rows, D = 16x16 fp32 accumulator held in VGPRs.
// 16 waves/workgroup cover the 256 output columns (16 cols per wave).
//
// Main loop is guard-free (16 nnz = 4 WMMAs per iteration) so the COO
// scalar reads use constant offsets (mergeable s_loads, no branches); the
// tail uses clamped addresses + selects instead of conditional loads.

typedef __attribute__((ext_vector_type(2))) float v2f;
typedef __attribute__((ext_vector_type(8))) float v8f;

#define OUTF 256
#define ROWS_PER_BLOCK 16

__device__ __forceinline__ int lower_bound_i32(const int* __restrict__ arr,
                                               int n, int key) {
  int lo = 0, hi = n;
  while (lo < hi) {
    int mid = (lo + hi) >> 1;
    if (arr[mid] < key) lo = mid + 1; else hi = mid;
  }
  return lo;
}

__global__ __launch_bounds__(512) void
spmm_wmma_f32_kernel(const int* __restrict__ row_idx,
                     const int* __restrict__ col_idx,
                     const float* __restrict__ vals,
                     const float* __restrict__ weight,
                     const float* __restrict__ bias,
                     float* __restrict__ out, int nnz) {
  const int rbase = blockIdx.x * ROWS_PER_BLOCK;   // first of 16 output rows
  const int tid   = threadIdx.x;
  const int wave  = tid >> 5;        // 0..15 -> which 16-col N tile
  const int lane  = tid & 31;
  const int laneM = lane & 15;       // M index (A) / N index (B, C, D)
  const int half  = lane >> 4;       // 0: lanes 0-15, 1: lanes 16-31

  const int nbase = wave * 16;       // base output column for this wave

  // Nonzero range for this 16-row slab (uniform across the workgroup).
  const int start = lower_bound_i32(row_idx, nnz, rbase);
  const int end   = lower_bound_i32(row_idx, nnz, rbase + ROWS_PER_BLOCK);

  v8f acc = {0.f, 0.f, 0.f, 0.f, 0.f, 0.f, 0.f, 0.f};

  int k = start;
  const int full_end = start + ((end - start) & ~15);

  // ---------------- Main loop: 16 nnz = 4 WMMAs, no guards ----------------
  for (; k < full_end; k += 16) {
    // Stream-prefetch the COO arrays ahead (emits global_prefetch_b8).
    __builtin_prefetch(&vals[k + 256], 0, 0);
    __builtin_prefetch(&col_idx[k + 256], 0, 0);
    __builtin_prefetch(&row_idx[k + 256], 0, 0);

    int   rr[16], cc[16];
    float vv[16];
#pragma unroll
    for (int j = 0; j < 16; ++j) {
      rr[j] = row_idx[k + j] - rbase;   // uniform -> s_load, const offsets
      cc[j] = col_idx[k + j];
      vv[j] = vals[k + j];
    }

#pragma unroll
    for (int j = 0; j < 4; ++j) {
      const int q = 4 * j;
      // A tile 16x4 fp32 (one-hot): VGPR0 = K0|K2 halves, VGPR1 = K1|K3.
      const int   rlo = half ? rr[q + 2] : rr[q + 0];
      const float vlo = half ? vv[q + 2] : vv[q + 0];
      const int   rhi = half ? rr[q + 3] : rr[q + 1];
      const float vhi = half ? vv[q + 3] : vv[q + 1];
      v2f a;
      a.x = (rlo == laneM) ? vlo : 0.f;
      a.y = (rhi == laneM) ? vhi : 0.f;

      // B tile 4x16 fp32: two coalesced 16-lane gathers of weight rows.
      const int clo = half ? cc[q + 2] : cc[q + 0];
      const int chi = half ? cc[q + 3] : cc[q + 1];
      v2f b;
      b.x = weight[clo * OUTF + nbase + laneM];
      b.y = weight[chi * OUTF + nbase + laneM];

      // 8 args: (neg_a, A, neg_b, B, c_mod, C, reuse_a, reuse_b)
      acc = __builtin_amdgcn_wmma_f32_16x16x4_f32(
          false, a, false, b, (short)0, acc, false, false);
    }
  }

  // ---------------- Tail: <=15 nnz, branchless clamp + select -------------
  for (; k < end; k += 4) {
    const int last = end - 1;
    const int k1 = (k + 1 < end) ? k + 1 : last;   // safe addresses
    const int k2 = (k + 2 < end) ? k + 2 : last;
    const int k3 = (k + 3 < end) ? k + 3 : last;

    // Unconditional loads at clamped (valid) addresses, then select.
    int   r0 = row_idx[k]  - rbase;
    int   r1 = row_idx[k1] - rbase;
    int   r2 = row_idx[k2] - rbase;
    int   r3 = row_idx[k3] - rbase;
    float v0 = vals[k];
    float v1 = vals[k1];
    float v2 = vals[k2];
    float v3 = vals[k3];
    const int c0 = col_idx[k];
    const int c1 = col_idx[k1];
    const int c2 = col_idx[k2];
    const int c3 = col_idx[k3];

    r1 = (k + 1 < end) ? r1 : -1;  v1 = (k + 1 < end) ? v1 : 0.f;
    r2 = (k + 2 < end) ? r2 : -1;  v2 = (k + 2 < end) ? v2 : 0.f;
    r3 = (k + 3 < end) ? r3 : -1;  v3 = (k + 3 < end) ? v3 : 0.f;

    const int   rlo = half ? r2 : r0;
    const float vlo = half ? v2 : v0;
    const int   rhi = half ? r3 : r1;
    const float vhi = half ? v3 : v1;
    v2f a;
    a.x = (rlo == laneM) ? vlo : 0.f;
    a.y = (rhi == laneM) ? vhi : 0.f;

    const int clo = half ? c2 : c0;
    const int chi = half ? c3 : c1;
    v2f b;
    b.x = weight[clo * OUTF + nbase + laneM];
    b.y = weight[chi * OUTF + nbase + laneM];

    acc = __builtin_amdgcn_wmma_f32_16x16x4_f32(
        false, a, false, b, (short)0, acc, false, false);
  }

  // ---------------- Epilogue: bias + store --------------------------------
  // C/D layout: VGPR j holds (M=j, lanes 0-15) and (M=j+8, lanes 16-31).
  const float bv = bias[nbase + laneM];
#pragma unroll
  for (int j = 0; j < 8; ++j) {
    const int m = j + half * 8;
    out[(rbase + m) * OUTF + nbase + laneM] = acc[j] + bv;
  }
}

extern "C" void kernel_launch(void* const* d_in, const int* in_sizes, int n_in,
                              void* d_out, int out_size, void* d_ws,
                              size_t ws_size, hipStream_t stream) {
  const int*   row_idx = (const int*)d_in[0];
  const int*   col_idx = (const int*)d_in[1];
  const float* vals    = (const float*)d_in[2];
  const float* weight  = (const float*)d_in[3];
  const float* bias    = (const float*)d_in[4];
  float*       out     = (float*)d_out;

  const int nnz     = in_sizes[0];
  const int batch   = out_size / OUTF;           // 4096
  const int nblocks = batch / ROWS_PER_BLOCK;    // 256

  spmm_wmma_f32_kernel<<<dim3(nblocks), dim3(512), 0, stream>>>(
      row_idx, col_idx, vals, weight, bias, out, nnz);
}